// BipartiteMatchingLoss_90348932039003
// MI455X (gfx1250) — compile-verified
//
#include <hip/hip_runtime.h>
#include <hip/hip_bf16.h>
#include <math.h>

// Problem sizes (from setup_inputs): B=128, S=20, C=32000
#define BB   128
#define SS   20
#define CC   32000
#define ROWS (BB * SS)        // 2560

// ---- logsumexp kernel config ----
#define TPB    256
#define NWAVE  8              // 256 / 32
#define CHUNKF 4096           // floats per staged chunk (16 KB)
#define NCHUNK 8              // ceil(32000 / 4096); last chunk = 3328 real floats

typedef float v8f __attribute__((ext_vector_type(8)));
typedef float v2f __attribute__((ext_vector_type(2)));
typedef int   v4i_vs __attribute__((vector_size(16)));   // matches builtin param type

#if defined(__gfx1250__) && __has_builtin(__builtin_amdgcn_global_load_async_to_lds_b128) && __has_builtin(__builtin_amdgcn_s_wait_asynccnt)
#define HAVE_ASYNC 1
#else
#define HAVE_ASYNC 0
#endif

#if defined(__gfx1250__) && __has_builtin(__builtin_amdgcn_wmma_f32_16x16x4_f32)
#define HAVE_WMMA4 1
#else
#define HAVE_WMMA4 0
#endif

// ---------------- wave32 reduction helpers ----------------
__device__ __forceinline__ float wred_maxf(float v) {
  #pragma unroll
  for (int o = 16; o; o >>= 1) v = fmaxf(v, __shfl_xor(v, o, 32));
  return v;
}
__device__ __forceinline__ float wred_sumf(float v) {
  #pragma unroll
  for (int o = 16; o; o >>= 1) v += __shfl_xor(v, o, 32);
  return v;
}

// ---------------- async staging of one chunk into LDS ----------------
// Deterministic ASYNCcnt accounting: exactly 4 async B128 instructions per wave
// per chunk (addresses clamped, never masked off, so every wave issues all 4).
__device__ __forceinline__ void stage_chunk(const float* __restrict__ src,
                                            float* bufp, int k, int tid) {
  const int base = k * CHUNKF;
  const int nf   = (CC - base) < CHUNKF ? (CC - base) : CHUNKF; // multiple of 4
  const int nB   = nf >> 2;                                     // B128 packets
#if HAVE_ASYNC
  #pragma unroll
  for (int it = 0; it < 4; ++it) {
    int c  = tid + it * TPB;
    int cc = c < nB ? c : (nB - 1);   // clamp: duplicate loads are harmless
    __builtin_amdgcn_global_load_async_to_lds_b128(
        (v4i_vs*)(src + base + 4 * cc),
        (v4i_vs*)(bufp + 4 * cc),
        0, 0);
  }
#else
  #pragma unroll
  for (int it = 0; it < 4; ++it) {
    int c  = tid + it * TPB;
    int cc = c < nB ? c : (nB - 1);
    *(float4*)(bufp + 4 * cc) = *(const float4*)(src + base + 4 * cc);
  }
#endif
  // pad tail of last chunk with a very negative value: max() unaffected by
  // real data, exp(pad - M) underflows to exactly 0.
  for (int i = nf + tid; i < CHUNKF; i += TPB) bufp[i] = -1e30f;
}

__device__ __forceinline__ void wait_async_le(int n) {
#if HAVE_ASYNC
  if (n > 0) __builtin_amdgcn_s_wait_asynccnt(4);
  else       __builtin_amdgcn_s_wait_asynccnt(0);
#else
  (void)n;
#endif
}

// ---------------- Kernel A: per-row logsumexp over C=32000 ----------------
// One block per (b,s) row. Async double-buffered LDS staging; online (running
// max) logsumexp; exp-sums accumulated on the matrix pipe via
// V_WMMA_F32_16X16X4_F32 with an all-ones B (sum of all D elements = 16 * sum
// of all A inputs, independent of VGPR layout details). Two independent C
// accumulators break the WMMA->WMMA RAW chain so the matrix pipe overlaps the
// v_exp_f32 stream instead of stalling on hazard NOPs.
__global__ __launch_bounds__(TPB) void lse_kernel(const float* __restrict__ outp,
                                                  float* __restrict__ lse) {
  __shared__ __align__(16) float buf[2][CHUNKF];
  __shared__ float red_m[NWAVE], red_s[NWAVE];

  const int row  = blockIdx.x;
  const float* src = outp + (size_t)row * CC;
  const int tid  = threadIdx.x;
  const int lane = tid & 31;
  const int wv   = tid >> 5;

  stage_chunk(src, buf[0], 0, tid);
  stage_chunk(src, buf[1], 1, tid);

  float Mrun = -INFINITY;
#if HAVE_WMMA4
  v8f Cacc0 = {0.f, 0.f, 0.f, 0.f, 0.f, 0.f, 0.f, 0.f};
  v8f Cacc1 = {0.f, 0.f, 0.f, 0.f, 0.f, 0.f, 0.f, 0.f};
  v2f Bones; Bones[0] = 1.0f; Bones[1] = 1.0f;
#else
  float ss = 0.f;
#endif

  for (int k = 0; k < NCHUNK; ++k) {
    // chunk k complete when at most the NEXT chunk's 4 loads are outstanding
    wait_async_le((k < NCHUNK - 1) ? 4 : 0);
    __syncthreads();

    const float* cb = buf[k & 1] + wv * (CHUNKF / NWAVE);  // 512 floats / wave

    // wave-local chunk max
    float lm = -INFINITY;
    #pragma unroll
    for (int i = 0; i < 16; ++i) lm = fmaxf(lm, cb[lane + i * 32]);
    float cm = wred_maxf(lm);
    float Mn = fmaxf(Mrun, cm);
    float sc = (Mrun == -INFINITY) ? 0.f : __expf(Mrun - Mn);
    Mrun = Mn;

#if HAVE_WMMA4
    #pragma unroll
    for (int r = 0; r < 8; ++r) { Cacc0[r] *= sc; Cacc1[r] *= sc; }
    #pragma unroll
    for (int g = 0; g < 8; g += 2) {
      float x0 = cb[g * 64 + lane * 2];
      float x1 = cb[g * 64 + lane * 2 + 1];
      float y0 = cb[(g + 1) * 64 + lane * 2];
      float y1 = cb[(g + 1) * 64 + lane * 2 + 1];
      v2f a0; a0[0] = __expf(x0 - Mn); a0[1] = __expf(x1 - Mn);
      v2f a1; a1[0] = __expf(y0 - Mn); a1[1] = __expf(y1 - Mn);
      Cacc0 = __builtin_amdgcn_wmma_f32_16x16x4_f32(
          false, a0, false, Bones, (short)0, Cacc0, false, false);
      Cacc1 = __builtin_amdgcn_wmma_f32_16x16x4_f32(
          false, a1, false, Bones, (short)0, Cacc1, false, false);
    }
#else
    ss *= sc;
    #pragma unroll
    for (int i = 0; i < 16; ++i) ss += __expf(cb[lane + i * 32] - Mn);
#endif
    __syncthreads();
    if (k + 2 < NCHUNK) stage_chunk(src, buf[k & 1], k + 2, tid);
  }

#if HAVE_WMMA4
  float t = 0.f;
  #pragma unroll
  for (int r = 0; r < 8; ++r) t += Cacc0[r] + Cacc1[r];
  float sw = wred_sumf(t) * (1.0f / 16.0f);  // sum(D) == 16 * sum(A)
#else
  float sw = wred_sumf(ss);
#endif
  if (lane == 0) { red_m[wv] = Mrun; red_s[wv] = sw; }
  __syncthreads();
  if (tid == 0) {
    float gm = red_m[0];
    #pragma unroll
    for (int w = 1; w < NWAVE; ++w) gm = fmaxf(gm, red_m[w]);
    float S = 0.f;
    #pragma unroll
    for (int w = 0; w < NWAVE; ++w) S += red_s[w] * __expf(red_m[w] - gm);
    lse[row] = gm + logf(S);
  }
}

// ---------------- Kernel B: restricted-softmax cost matrix ----------------
// cost[b][i][j] = -log_softmax_j(outputs[b,i,gold[b,j]]) * weight[gold[b,j]]
// One wave per (b,i) row; lanes 0..19 = columns j.
__global__ __launch_bounds__(256) void cost_kernel(const float* __restrict__ outp,
                                                   const int* __restrict__ gold,
                                                   const float* __restrict__ weight,
                                                   float* __restrict__ cost) {
  const int tid  = threadIdx.x;
  const int lane = tid & 31;
  const int wv   = tid >> 5;
  const int row  = blockIdx.x * 8 + wv;          // (b*S + i)
  const int b    = row / SS;

  float val = -INFINITY, w = 0.f;
  if (lane < SS) {
    int g = gold[b * SS + lane];
    val   = outp[(size_t)row * CC + g];
    w     = weight[g];
  }
  float mx = wred_maxf(val);
  float e  = (lane < SS) ? __expf(val - mx) : 0.f;
  float s  = wred_sumf(e);
  float ls = logf(s);
  if (lane < SS) {
    float c = -(val - mx - ls) * w;
    if (isinf(c)) c = 1e8f;                      // BIG, as in the reference
    cost[(size_t)row * SS + lane] = c;
  }
}

// ---------------- Kernel C: Hungarian (shortest augmenting path) ----------
// One wave (32 threads) per batch element; lanes parallelize the m+1=21
// column operations. Potentials/deltas in double to match the reference's
// float64 decision process.
__global__ void hungarian_kernel(const float* __restrict__ cost,
                                 const int* __restrict__ gold,
                                 int* __restrict__ match) {
  const int b    = blockIdx.x;
  const int lane = threadIdx.x;                  // 32 threads
  const float* Cb = cost + (size_t)b * SS * SS;

  __shared__ double u_sh[SS + 1];
  __shared__ int    p_sh[SS + 1];
  if (lane <= SS) { u_sh[lane] = 0.0; p_sh[lane] = 0; }
  double v_r = 0.0;                              // v[lane]
  __syncthreads();

  for (int i = 1; i <= SS; ++i) {
    if (lane == 0) p_sh[0] = i;
    __syncthreads();
    double minv_r = INFINITY;
    int    way_r  = 0;
    bool   used_r = false;
    int j0 = 0;
    while (true) {
      if (lane == j0) used_r = true;
      const int i0     = p_sh[j0];
      const double ui0 = u_sh[i0];
      const bool actc  = (lane >= 1 && lane <= SS);
      if (actc && !used_r) {
        double cur = (double)Cb[(i0 - 1) * SS + (lane - 1)] - ui0 - v_r;
        if (cur < minv_r) { minv_r = cur; way_r = j0; }
      }
      double cand = (actc && !used_r) ? minv_r : INFINITY;
      // wave argmin, first index wins ties (matches np.argmin)
      double bv = cand; int bi = lane;
      #pragma unroll
      for (int o = 16; o; o >>= 1) {
        double ov = __shfl_xor(bv, o, 32);
        int    oi = __shfl_xor(bi, o, 32);
        if (ov < bv || (ov == bv && oi < bi)) { bv = ov; bi = oi; }
      }
      const int    j1    = bi;
      const double delta = bv;
      if (used_r && lane <= SS) {
        int pj = p_sh[lane];                     // distinct rows across used lanes
        u_sh[pj] += delta;
        v_r      -= delta;
      }
      if (actc && !used_r) minv_r -= delta;
      __syncthreads();
      j0 = j1;
      if (p_sh[j0] == 0) break;
    }
    // augment along 'way' chain
    while (j0 != 0) {
      int j1a = __shfl(way_r, j0, 32);
      int pj1 = p_sh[j1a];
      __syncthreads();
      if (lane == 0) p_sh[j0] = pj1;
      __syncthreads();
      j0 = j1a;
    }
    __syncthreads();
  }
  if (lane >= 1 && lane <= SS) {
    int pj = p_sh[lane];
    if (pj != 0) match[b * SS + (pj - 1)] = gold[b * SS + (lane - 1)];
  }
}

// ---------------- Kernel D: final masked-mean NLL ----------------
__global__ __launch_bounds__(256) void loss_kernel(const float* __restrict__ outp,
                                                   const unsigned char* __restrict__ mask,
                                                   const float* __restrict__ weight,
                                                   const float* __restrict__ lse,
                                                   const int* __restrict__ match,
                                                   float* __restrict__ res) {
  __shared__ float s_s[256], s_c[256];
  const int tid = threadIdx.x;
  float s = 0.f, c = 0.f;
  for (int idx = tid; idx < ROWS; idx += 256) {
    float m  = mask[idx] ? 1.f : 0.f;
    int   mt = match[idx];
    float x  = outp[(size_t)idx * CC + mt];
    float nll = (lse[idx] - x) * weight[mt];
    s += nll * m;
    c += m;
  }
  s_s[tid] = s; s_c[tid] = c;
  __syncthreads();
  for (int o = 128; o; o >>= 1) {
    if (tid < o) { s_s[tid] += s_s[tid + o]; s_c[tid] += s_c[tid + o]; }
    __syncthreads();
  }
  if (tid == 0) res[0] = s_s[0] / (s_c[0] + 1e-8f);
}

// ---------------- launcher ----------------
extern "C" void kernel_launch(void* const* d_in, const int* in_sizes, int n_in,
                              void* d_out, int out_size, void* d_ws, size_t ws_size,
                              hipStream_t stream) {
  const float*         outputs = (const float*)d_in[0];
  const int*           gold    = (const int*)d_in[1];          // int32 labels
  const unsigned char* mask    = (const unsigned char*)d_in[2]; // jnp bool = 1 byte
  const float*         weight  = (const float*)d_in[3];
  float*               res     = (float*)d_out;

  char*  ws    = (char*)d_ws;
  float* lse   = (float*)ws;                                   // ROWS floats
  float* cost  = (float*)(ws + (size_t)ROWS * 4);              // ROWS*SS floats
  int*   match = (int*)  (ws + (size_t)ROWS * 4 + (size_t)ROWS * SS * 4);

  lse_kernel      <<<ROWS,     TPB, 0, stream>>>(outputs, lse);
  cost_kernel     <<<ROWS / 8, 256, 0, stream>>>(outputs, gold, weight, cost);
  hungarian_kernel<<<BB,       32,  0, stream>>>(cost, gold, match);
  loss_kernel     <<<1,        256, 0, stream>>>(outputs, mask, weight, lse, match, res);
}